// DeformConv2d_81664508166832
// MI455X (gfx1250) — compile-verified
//
#include <hip/hip_runtime.h>
#include <hip/hip_bf16.h>
#include <math.h>

// DCNv2 forward for MI455X (gfx1250, wave32).
// Stage 1: per-pixel offset/modulation convs -> sampling metadata (4 idx + 4 wgt per (pixel,n))
// Stage 2: fused bilinear-gather + GEMM using V_WMMA_F32_16X16X4_F32 (fp32 exact, K=4
//          matches the rate at which gathers can produce the B operand).

typedef float v2f __attribute__((ext_vector_type(2)));
typedef float v4f __attribute__((ext_vector_type(4)));
typedef int   v4i __attribute__((ext_vector_type(4)));
typedef float v8f __attribute__((ext_vector_type(8)));

#define BSZ   8
#define CCH   64
#define HH    96
#define WW    96
#define OCH   64
#define NPTS  9
#define HW    (HH * WW)        // 9216
#define PIX   (BSZ * HW)       // 73728
#define KDIM  (CCH * NPTS)     // 576

// ---------------------------------------------------------------------------
// Kernel 1: offset/modulation convs + sampling metadata.
// One thread per output pixel; computes all 27 conv channels (x loaded once,
// weights are block-uniform -> scalar-load friendly), then emits per (pixel,n):
//   meta[0..3] = bilinear corner weights * sigmoid(mod)  (0 if corner in zero-pad)
//   meta[4..7] = linear index into the unpadded 96x96 x-plane (0 if invalid)
// ---------------------------------------------------------------------------
__global__ __launch_bounds__(256) void dcn_meta_kernel(
    const float* __restrict__ x,        // [B][C][H][W]
    const float* __restrict__ offw,     // [18][C][3][3]
    const float* __restrict__ offb,     // [18]
    const float* __restrict__ mw,       // [9][C][3][3]
    const float* __restrict__ mb,       // [9]
    float* __restrict__ meta)           // [PIX][9][8]
{
    int pix = blockIdx.x * 256 + threadIdx.x;
    if (pix >= PIX) return;
    int b = pix / HW;
    int p = pix - b * HW;
    int h = p / WW;
    int w = p - h * WW;

    const float* xb = x + (size_t)b * CCH * HW;

    float sx[NPTS] = {}, sy[NPTS] = {}, sm[NPTS] = {};
    for (int c = 0; c < CCH; ++c) {
        const float* xc = xb + c * HW;
        #pragma unroll
        for (int kh = 0; kh < 3; ++kh) {
            int ih = h + kh - 1;
            bool okh = (unsigned)ih < (unsigned)HH;
            #pragma unroll
            for (int kw = 0; kw < 3; ++kw) {
                int iw = w + kw - 1;
                float xv = (okh && (unsigned)iw < (unsigned)WW) ? xc[ih * WW + iw] : 0.0f;
                int wi = c * 9 + kh * 3 + kw;
                #pragma unroll
                for (int n = 0; n < NPTS; ++n) {
                    sx[n] = fmaf(xv, offw[(n         * CCH * 9) + wi], sx[n]);
                    sy[n] = fmaf(xv, offw[((n + NPTS) * CCH * 9) + wi], sy[n]);
                    sm[n] = fmaf(xv, mw  [(n         * CCH * 9) + wi], sm[n]);
                }
            }
        }
    }

    #pragma unroll
    for (int n = 0; n < NPTS; ++n) {
        float offx = sx[n] + offb[n];
        float offy = sy[n] + offb[n + NPTS];
        float mod  = 1.0f / (1.0f + __expf(-(sm[n] + mb[n])));

        // padded coords: Hp = Wp = 98, center grid = (h+1, w+1), taps in {-1,0,1}
        float px = offx + (float)(h + 1) + (float)(n / 3 - 1);
        float py = offy + (float)(w + 1) + (float)(n % 3 - 1);

        float flx = floorf(px), fly = floorf(py);
        float tlx = fminf(fmaxf(flx,        0.0f), 97.0f);
        float tly = fminf(fmaxf(fly,        0.0f), 97.0f);
        float brx = fminf(fmaxf(flx + 1.0f, 0.0f), 97.0f);
        float bry = fminf(fmaxf(fly + 1.0f, 0.0f), 97.0f);
        float pcx = fminf(fmaxf(px,         0.0f), 97.0f);
        float pcy = fminf(fmaxf(py,         0.0f), 97.0f);

        float gx_t = 1.0f + tlx - pcx;
        float gx_b = 1.0f - (brx - pcx);
        float gy_l = 1.0f + tly - pcy;
        float gy_r = 1.0f - (bry - pcy);

        int ux[4] = {(int)tlx, (int)tlx, (int)brx, (int)brx};
        int uy[4] = {(int)tly, (int)bry, (int)tly, (int)bry};
        float gg[4] = {gx_t * gy_l, gx_t * gy_r, gx_b * gy_l, gx_b * gy_r};

        float* mp = meta + ((size_t)pix * NPTS + n) * 8;
        int*   mi = (int*)mp;
        #pragma unroll
        for (int q = 0; q < 4; ++q) {
            int u = ux[q], v = uy[q];
            // padded border (u==0||u==97||v==0||v==97) samples zero-pad -> weight 0
            bool valid = (u >= 1) & (u <= HH) & (v >= 1) & (v <= WW);
            mp[q]     = valid ? gg[q] * mod : 0.0f;
            mi[4 + q] = valid ? ((u - 1) * WW + (v - 1)) : 0;
        }
    }
}

// ---------------------------------------------------------------------------
// Kernel 2: fused bilinear sampling + GEMM with f32 WMMA.
// Block = 128 threads (4 wave32) = 64 pixels of one batch image.
// Wave tile: D[16 out-ch][16 pixels]; 4 out-ch tiles per wave so each gathered
// B fragment (sampled x_p) feeds 4 WMMAs. A = conv_w rows (natural [O][C*N]
// layout, no transpose needed). K = 576 -> 144 x v_wmma_f32_16x16x4_f32.
// D layout makes stores 16-lane contiguous over pixels (coalesced).
// ---------------------------------------------------------------------------
__global__ __launch_bounds__(128) void dcn_gemm_kernel(
    const float* __restrict__ x,        // [B][C][9216]
    const float* __restrict__ conv_w,   // [64][576]
    const float* __restrict__ conv_b,   // [64]
    const float* __restrict__ meta,     // [PIX][9][8]
    float* __restrict__ out)            // [B][64][9216]
{
    __shared__ float smeta[64 * NPTS * 8];   // 18 KB: metadata for this block's 64 pixels

    int tid = threadIdx.x;
    int block_pix0 = blockIdx.x * 64;        // global pixel base (batch-aligned: 9216 % 64 == 0)

    // cooperative coalesced stage of metadata into LDS (4608 dwords)
    const float* msrc = meta + (size_t)block_pix0 * NPTS * 8;
    for (int i = tid; i < 64 * NPTS * 8; i += 128)
        smeta[i] = msrc[i];
    __syncthreads();

    int wave = tid >> 5;
    int lane = tid & 31;
    int col  = lane & 15;                    // N index: pixel-in-wave / M index for A rows
    int hi   = lane >> 4;                    // 0: K rows {k0,k0+1}; 1: K rows {k0+2,k0+3}
    int lpix = (wave << 4) + col;            // pixel within block (0..63)

    int b     = block_pix0 / HW;
    int pbase = block_pix0 - b * HW;         // pixel-in-batch base

    const float* xb = x + (size_t)b * CCH * HW;

    // A row pointers: o = t*16 + col, row length KDIM
    const float* ar0 = conv_w + (size_t)(0 * 16 + col) * KDIM;
    const float* ar1 = conv_w + (size_t)(1 * 16 + col) * KDIM;
    const float* ar2 = conv_w + (size_t)(2 * 16 + col) * KDIM;
    const float* ar3 = conv_w + (size_t)(3 * 16 + col) * KDIM;

    v8f acc0 = {}, acc1 = {}, acc2 = {}, acc3 = {};

    // k = c*9 + n; this lane handles kk = k0 + 2*hi and kk+1. Track (c,n) incrementally.
    int c0 = 0;
    int n0 = hi << 1;                        // 0 or 2
    for (int k0 = 0; k0 < KDIM; k0 += 4) {
        int kcur = k0 + (n0 - (c0 * 0)) * 0 + (hi << 1);   // = k0 + 2*hi
        int n_a = n0, c_a = c0;
        int n_b = n_a + 1, c_b = c_a;
        if (n_b >= NPTS) { n_b -= NPTS; c_b += 1; }

        // ---- B fragment: two bilinear samples for this lane's pixel ----
        const float* ma = &smeta[((lpix * NPTS) + n_a) * 8];
        v4f wa = *(const v4f*)ma;
        v4i ia = *(const v4i*)(ma + 4);
        const float* xc_a = xb + c_a * HW;
        float bv0 = wa[0] * xc_a[ia[0]] + wa[1] * xc_a[ia[1]]
                  + wa[2] * xc_a[ia[2]] + wa[3] * xc_a[ia[3]];

        const float* mbp = &smeta[((lpix * NPTS) + n_b) * 8];
        v4f wb = *(const v4f*)mbp;
        v4i ib = *(const v4i*)(mbp + 4);
        const float* xc_b = xb + c_b * HW;
        float bv1 = wb[0] * xc_b[ib[0]] + wb[1] * xc_b[ib[1]]
                  + wb[2] * xc_b[ib[2]] + wb[3] * xc_b[ib[3]];

        v2f bfrag; bfrag[0] = bv0; bfrag[1] = bv1;

        // ---- A fragments (conv_w is small + hot in L2; b64 loads, kcur is even) ----
        v2f a0; a0[0] = ar0[kcur]; a0[1] = ar0[kcur + 1];
        v2f a1; a1[0] = ar1[kcur]; a1[1] = ar1[kcur + 1];
        v2f a2; a2[0] = ar2[kcur]; a2[1] = ar2[kcur + 1];
        v2f a3; a3[0] = ar3[kcur]; a3[1] = ar3[kcur + 1];

        // D = A(16x4) * B(4x16) + C  — fp32 WMMA, exact reference precision
        acc0 = __builtin_amdgcn_wmma_f32_16x16x4_f32(false, a0, false, bfrag, (short)0, acc0, false, false);
        acc1 = __builtin_amdgcn_wmma_f32_16x16x4_f32(false, a1, false, bfrag, (short)0, acc1, false, false);
        acc2 = __builtin_amdgcn_wmma_f32_16x16x4_f32(false, a2, false, bfrag, (short)0, acc2, false, false);
        acc3 = __builtin_amdgcn_wmma_f32_16x16x4_f32(false, a3, false, bfrag, (short)0, acc3, false, false);

        // advance (c,n) by 4 (single wrap since 4 < 9)
        n0 += 4;
        if (n0 >= NPTS) { n0 -= NPTS; c0 += 1; }
    }

    // ---- Epilogue: D[m][n] -> lane = n + 16*(m>=8), vgpr = m&7.
    // m = out-ch in tile, n = pixel -> contiguous-pixel stores across lanes.
    int  m_off = hi << 3;
    int  pout  = pbase + (wave << 4) + col;
    float* ob  = out + (size_t)b * OCH * HW + pout;
    #pragma unroll
    for (int r = 0; r < 8; ++r) {
        int mo = r + m_off;                  // out-ch within 16-wide tile
        ob[(size_t)(0 * 16 + mo) * HW] = acc0[r] + conv_b[0 * 16 + mo];
        ob[(size_t)(1 * 16 + mo) * HW] = acc1[r] + conv_b[1 * 16 + mo];
        ob[(size_t)(2 * 16 + mo) * HW] = acc2[r] + conv_b[2 * 16 + mo];
        ob[(size_t)(3 * 16 + mo) * HW] = acc3[r] + conv_b[3 * 16 + mo];
    }
}

// ---------------------------------------------------------------------------
extern "C" void kernel_launch(void* const* d_in, const int* in_sizes, int n_in,
                              void* d_out, int out_size, void* d_ws, size_t ws_size,
                              hipStream_t stream) {
    const float* x    = (const float*)d_in[0];   // [8,64,96,96]
    const float* offw = (const float*)d_in[1];   // [18,64,3,3]
    const float* offb = (const float*)d_in[2];   // [18]
    const float* mw   = (const float*)d_in[3];   // [9,64,3,3]
    const float* mb   = (const float*)d_in[4];   // [9]
    const float* cw   = (const float*)d_in[5];   // [64,64,9] -> [64][576]
    const float* cb   = (const float*)d_in[6];   // [64]
    float* out  = (float*)d_out;                 // [8,64,96,96]
    float* meta = (float*)d_ws;                  // needs PIX*9*8*4 = 21.2 MB scratch

    dcn_meta_kernel<<<PIX / 256, 256, 0, stream>>>(x, offw, offb, mw, mb, meta);
    dcn_gemm_kernel<<<PIX / 64, 128, 0, stream>>>(x, cw, cb, meta, out);
}